// ChebConv_83133386981495
// MI455X (gfx1250) — compile-verified
//
#include <hip/hip_runtime.h>

// ChebConv fused kernel for MI455X (gfx1250, wave32, WMMA).
//
// Shapes: B=2, FIN=32, V=768, SP=X*Y*Z=512, K=5, FOUT=32, NNZ=6144.
// Memory-bound problem (~10 GFLOP vs ~200MB minimal traffic @ 23.3 TB/s).
// Strategy: one workgroup per (b, spatial) slice; entire Chebyshev recurrence
// lives in LDS ([768 x 16] f32, two fin-halves, ping-pong buffers with the
// "negate + 2*scatter" trick), GEMM contraction done with v_wmma_f32_16x16x4_f32
// accumulating across all K orders and both fin halves in registers.

#define VV     768
#define FINC   32
#define FOUTC  32
#define KCH    5
#define SPC    512
#define NNZC   6144
#define HALF_F 16
#define PSTR   17      // padded LDS row stride (floats): conflict-free banks
#define BQ     2

typedef float v2f __attribute__((ext_vector_type(2)));
typedef float v8f __attribute__((ext_vector_type(8)));

// ---------------------------------------------------------------------------
// Prepass: x [B][FIN][V][SP]  ->  xt [B][SP][V][FIN]  (coalesced both sides)
// ---------------------------------------------------------------------------
__global__ __launch_bounds__(256) void cheb_transpose(
    const float* __restrict__ x, float* __restrict__ xt)
{
    __shared__ float tile[32][33];
    const int blk  = blockIdx.x;          // BQ * VV * (SPC/32) blocks
    const int spb  = blk & 15;
    const int rest = blk >> 4;
    const int v    = rest % VV;
    const int b    = rest / VV;
    const int sp0  = spb * 32;

    const int t    = threadIdx.x;
    const int col  = t & 31;              // sp offset on load, fin on store
    const int rowb = t >> 5;              // 0..7

#pragma unroll
    for (int i = 0; i < 4; ++i) {
        const int fin = i * 8 + rowb;     // 0..31
        tile[fin][col] =
            x[(((size_t)b * FINC + fin) * VV + v) * SPC + sp0 + col];
    }
    __syncthreads();
#pragma unroll
    for (int i = 0; i < 4; ++i) {
        const int sp = i * 8 + rowb;      // 0..31
        xt[(((size_t)b * SPC + sp0 + sp) * VV + v) * FINC + col] = tile[col][sp];
    }
}

// ---------------------------------------------------------------------------
// GEMM step: acc[v_tile][n_tile] += T[768 x 16fin] * W_k[16fin x 32fout]
// using v_wmma_f32_16x16x4_f32.  Per the ISA 32-bit operand layouts:
//   A 16x4:  lanes 0-15 -> K=0,1 in the 2 VGPRs; lanes 16-31 -> K=2,3
//   B 4x16:  VGPR0 = K-row (0 | 2), VGPR1 = K-row (1 | 3), N = lane%16
//   C/D:     VGPR g -> rows g (lanes 0-15) and g+8 (lanes 16-31), N = lane%16
// ---------------------------------------------------------------------------
__device__ __forceinline__ void gemm_step(
    v8f acc[6][2], const float* __restrict__ Tb,
    const float* __restrict__ wk,   // = wl + (k*FINC + f0)*FOUTC
    int m0, int ll, int hl)
{
#pragma unroll
    for (int n = 0; n < 2; ++n) {
#pragma unroll
        for (int kk = 0; kk < 4; ++kk) {
            const float* wb = wk + (kk * 4 + 2 * hl) * FOUTC + n * 16 + ll;
            v2f bf;
            bf.x = wb[0];
            bf.y = wb[FOUTC];
#pragma unroll
            for (int j = 0; j < 6; ++j) {
                const float* ap =
                    Tb + ((m0 + j) * 16 + ll) * PSTR + kk * 4 + 2 * hl;
                v2f af;
                af.x = ap[0];
                af.y = ap[1];
                acc[j][n] = __builtin_amdgcn_wmma_f32_16x16x4_f32(
                    false, af, false, bf, (short)0, acc[j][n], false, false);
            }
        }
    }
}

// ---------------------------------------------------------------------------
// Main fused kernel: one workgroup (256 threads = 8 waves) per (b, sp) slice.
// ---------------------------------------------------------------------------
__global__ __launch_bounds__(256) void cheb_main(
    const int*   __restrict__ rows, const int* __restrict__ cols,
    const float* __restrict__ vals,
    const float* __restrict__ xsrc,          // original layout
    const float* __restrict__ xt, int use_xt,// transposed layout (optional)
    const float* __restrict__ weight, const float* __restrict__ bias,
    float* __restrict__ out)
{
    extern __shared__ float smem[];
    float* bufA = smem;                         // 768*17 floats
    float* bufB = bufA + VV * PSTR;             // 768*17 floats
    float* wl   = bufB + VV * PSTR;             // 5*32*32 floats
    float* bl   = wl + KCH * FINC * FOUTC;      // 32 floats

    const int t  = threadIdx.x;
    const int wg = blockIdx.x;                  // 0..1023
    const int b  = wg >> 9;
    const int sp = wg & (SPC - 1);

    // stage weights + bias once
    for (int i = t; i < KCH * FINC * FOUTC; i += 256) wl[i] = weight[i];
    if (t < FOUTC) bl[t] = bias[t];

    const int lane = t & 31;
    const int wave = t >> 5;
    const int hl   = lane >> 4;                 // upper half of wave
    const int ll   = lane & 15;
    const int m0   = wave * 6;                  // 48 v-tiles / 8 waves

    v8f acc[6][2];
#pragma unroll
    for (int j = 0; j < 6; ++j)
#pragma unroll
        for (int n = 0; n < 2; ++n)
            acc[j][n] = (v8f){0.f, 0.f, 0.f, 0.f, 0.f, 0.f, 0.f, 0.f};

    const float* xb_t = xt + (size_t)wg * (VV * FINC);

#pragma unroll 1
    for (int h = 0; h < 2; ++h) {               // fin halves
        const int f0 = h * HALF_F;
        __syncthreads();                        // bufA free to overwrite

        // ---- load T0 slice [768 x 16] into bufA ----
        if (use_xt) {
            for (int i = t; i < VV * HALF_F; i += 256) {
                const int v = i >> 4, f = i & (HALF_F - 1);
                bufA[v * PSTR + f] = xb_t[v * FINC + f0 + f];
            }
        } else {
            for (int i = t; i < VV * HALF_F; i += 256) {
                const int v = i >> 4, f = i & (HALF_F - 1);
                bufA[v * PSTR + f] =
                    xsrc[(((size_t)b * FINC + f0 + f) * VV + v) * SPC + sp];
            }
        }
        __syncthreads();

        gemm_step(acc, bufA, wl + (0 * FINC + f0) * FOUTC, m0, ll, hl);

        float* src = bufA;
        float* dst = bufB;
#pragma unroll 1
        for (int k = 1; k < KCH; ++k) {
            // init: T1 = 0 + L*T0 ; Tk = -T(k-2) + 2*L*T(k-1)  (in-place)
            if (k == 1) {
                for (int i = t; i < VV * PSTR; i += 256) dst[i] = 0.f;
            } else {
                for (int i = t; i < VV * PSTR; i += 256) dst[i] = -dst[i];
            }
            __syncthreads();

            const float scale = (k == 1) ? 1.0f : 2.0f;
            // SpMM scatter: 6144 edges x 16 fins, ds_add_f32 atomics
            for (int i = t; i < NNZC * HALF_F; i += 256) {
                const int e = i >> 4, f = i & (HALF_F - 1);
                const float contrib =
                    scale * vals[e] * src[cols[e] * PSTR + f];
                atomicAdd(&dst[rows[e] * PSTR + f], contrib);
            }
            __syncthreads();

            gemm_step(acc, dst, wl + (k * FINC + f0) * FOUTC, m0, ll, hl);

            float* tmpp = src; src = dst; dst = tmpp;
        }
    }

    // ---- epilogue: + bias, scatter to out [B][FOUT][V][SP] ----
#pragma unroll
    for (int j = 0; j < 6; ++j) {
#pragma unroll
        for (int n = 0; n < 2; ++n) {
            const int fo = n * 16 + ll;
            const float bv = bl[fo];
            const size_t base = (((size_t)b * FOUTC + fo) * VV) * SPC + sp;
#pragma unroll
            for (int g = 0; g < 8; ++g) {
                const int v = (m0 + j) * 16 + g + 8 * hl;
                out[base + (size_t)v * SPC] = acc[j][n][g] + bv;
            }
        }
    }
}

// ---------------------------------------------------------------------------
extern "C" void kernel_launch(void* const* d_in, const int* in_sizes, int n_in,
                              void* d_out, int out_size, void* d_ws,
                              size_t ws_size, hipStream_t stream)
{
    const int*   rows = (const int*)d_in[0];
    const int*   cols = (const int*)d_in[1];
    const float* vals = (const float*)d_in[2];
    const float* x    = (const float*)d_in[3];
    const float* w    = (const float*)d_in[4];
    const float* bias = (const float*)d_in[5];
    float*       out  = (float*)d_out;

    const size_t xt_bytes = (size_t)BQ * SPC * VV * FINC * sizeof(float);
    const int use_xt = (ws_size >= xt_bytes) ? 1 : 0;
    float* xt = (float*)d_ws;

    if (use_xt) {
        cheb_transpose<<<BQ * VV * (SPC / 32), 256, 0, stream>>>(x, xt);
    }

    const size_t smem_bytes =
        (size_t)(2 * VV * PSTR + KCH * FINC * FOUTC + FOUTC) * sizeof(float);

    cheb_main<<<BQ * SPC, 256, smem_bytes, stream>>>(
        rows, cols, vals, x, xt, use_xt, w, bias, out);
}